// Pin2PinAttraction_14353780703797
// MI455X (gfx1250) — compile-verified
//
#include <hip/hip_runtime.h>

typedef float v2f __attribute__((ext_vector_type(2)));
typedef float v8f __attribute__((ext_vector_type(8)));
typedef int   v2i __attribute__((ext_vector_type(2)));

#define NBLOCKS  1024
#define NTHREADS 256   // 8 waves (wave32) per workgroup

// ---------------------------------------------------------------------------
// Kernel 1: grid-stride weighted squared-distance accumulation.
// Streams pairs/weights with non-temporal hints (read-once), gathers pin_pos
// through the cache hierarchy (8 MB -> L2 resident). Two independent
// accumulators per lane for ILP; they become the two A-matrix VGPRs of a
// V_WMMA_F32_16X16X4_F32 used as the 32-lane reduction.
// ---------------------------------------------------------------------------
__global__ __launch_bounds__(NTHREADS) void pin2pin_partial_kernel(
    const float* __restrict__ pin_pos,
    const float* __restrict__ weights,
    const int*   __restrict__ pairs,
    float*       __restrict__ partials,
    int n_pairs, int n_pins)
{
    const float* __restrict__ xs = pin_pos;
    const float* __restrict__ ys = pin_pos + n_pins;
    const v2i*   __restrict__ p2 = (const v2i*)pairs;

    float acc0 = 0.0f, acc1 = 0.0f;

    const int stride = gridDim.x * blockDim.x;
    int i = blockIdx.x * blockDim.x + threadIdx.x;

    // 2 pairs in flight per iteration -> more outstanding gathers per wave.
    for (; i + stride < n_pairs; i += 2 * stride) {
        v2i ab0 = __builtin_nontemporal_load(&p2[i]);
        v2i ab1 = __builtin_nontemporal_load(&p2[i + stride]);
        float w0 = __builtin_nontemporal_load(&weights[i]);
        float w1 = __builtin_nontemporal_load(&weights[i + stride]);

        float dx0 = xs[ab0.x] - xs[ab0.y];
        float dy0 = ys[ab0.x] - ys[ab0.y];
        float dx1 = xs[ab1.x] - xs[ab1.y];
        float dy1 = ys[ab1.x] - ys[ab1.y];

        acc0 = fmaf(w0, fmaf(dx0, dx0, dy0 * dy0), acc0);
        acc1 = fmaf(w1, fmaf(dx1, dx1, dy1 * dy1), acc1);
    }
    if (i < n_pairs) {
        v2i ab = __builtin_nontemporal_load(&p2[i]);
        float w = __builtin_nontemporal_load(&weights[i]);
        float dx = xs[ab.x] - xs[ab.y];
        float dy = ys[ab.x] - ys[ab.y];
        acc0 = fmaf(w, fmaf(dx, dx, dy * dy), acc0);
    }

    // ---- wave32 reduction via V_WMMA_F32_16X16X4_F32 ----
    // A (16x4 f32): lanes 0-15 -> K=0 (VGPR0=acc0), K=1 (VGPR1=acc1);
    //               lanes 16-31 -> K=2, K=3.
    // B (4x16) = all ones, C = 0.
    // D[m][n] = acc0[m]+acc1[m]+acc0[m+16]+acc1[m+16]  (same for every n).
    v2f a; a.x = acc0; a.y = acc1;
    v2f b; b.x = 1.0f; b.y = 1.0f;
    v8f c = {};
    c = __builtin_amdgcn_wmma_f32_16x16x4_f32(
        /*neg_a=*/false, a, /*neg_b=*/false, b,
        /*c_mod=*/(short)0, c, /*reuse_a=*/false, /*reuse_b=*/false);

    // Each lane sums its 8 D VGPRs: lanes 0-15 get rows 0-7, lanes 16-31 rows 8-15.
    float s = c[0] + c[1] + c[2] + c[3] + c[4] + c[5] + c[6] + c[7];
    // Combine the two half-wave row groups -> full wave sum in every lane.
    s += __shfl_xor(s, 16, 32);

    __shared__ float lds[NTHREADS / 32];
    const int lane = threadIdx.x & 31;
    const int wave = threadIdx.x >> 5;
    if (lane == 0) lds[wave] = s;
    __syncthreads();
    if (threadIdx.x == 0) {
        float t = 0.0f;
        #pragma unroll
        for (int wv = 0; wv < NTHREADS / 32; ++wv) t += lds[wv];
        partials[blockIdx.x] = t;
    }
}

// ---------------------------------------------------------------------------
// Kernel 2: single-block deterministic tree reduction of the block partials.
// Plain store to d_out (no atomics, no init dependence).
// ---------------------------------------------------------------------------
__global__ __launch_bounds__(1024) void pin2pin_final_kernel(
    const float* __restrict__ partials, float* __restrict__ out, int count)
{
    __shared__ float lds[32];
    float v = (threadIdx.x < (unsigned)count) ? partials[threadIdx.x] : 0.0f;

    #pragma unroll
    for (int off = 16; off > 0; off >>= 1) v += __shfl_down(v, off, 32);

    if ((threadIdx.x & 31) == 0) lds[threadIdx.x >> 5] = v;
    __syncthreads();

    if (threadIdx.x < 32) {
        float t = lds[threadIdx.x];
        #pragma unroll
        for (int off = 16; off > 0; off >>= 1) t += __shfl_down(t, off, 32);
        if (threadIdx.x == 0) out[0] = t;
    }
}

extern "C" void kernel_launch(void* const* d_in, const int* in_sizes, int n_in,
                              void* d_out, int out_size, void* d_ws, size_t ws_size,
                              hipStream_t stream)
{
    const float* pin_pos = (const float*)d_in[0];   // 2*NUM_PINS f32
    const float* weights = (const float*)d_in[1];   // NUM_PAIRS f32
    const int*   pairs   = (const int*)d_in[2];     // 2*NUM_PAIRS i32 (interleaved a,b)
    // d_in[3] = pin_mask (unused by the reference computation)

    const int n_pins  = in_sizes[0] / 2;
    const int n_pairs = in_sizes[1];

    // One f32 partial per block in workspace; clamp to available scratch
    // (deterministic for a given ws_size).
    int nblocks = NBLOCKS;
    if (ws_size < (size_t)NBLOCKS * sizeof(float)) {
        nblocks = (int)(ws_size / sizeof(float));
        if (nblocks < 1) nblocks = 1;
    }
    float* partials = (float*)d_ws;

    pin2pin_partial_kernel<<<nblocks, NTHREADS, 0, stream>>>(
        pin_pos, weights, pairs, partials, n_pairs, n_pins);
    pin2pin_final_kernel<<<1, 1024, 0, stream>>>(
        partials, (float*)d_out, nblocks);
}